// LGAN_18846316495223
// MI455X (gfx1250) — compile-verified
//
#include <hip/hip_runtime.h>
#include <math.h>

// ---------------------------------------------------------------------------
// Problem constants
// ---------------------------------------------------------------------------
#define BB 8
#define CC 63
#define HH 240
#define WW 240
#define HW (HH * WW)          // 57600
#define C2 126
#define WS 5
#define NWIN 48               // 240/5
#define BN_EPS 1e-5f

typedef __attribute__((ext_vector_type(2))) float v2f;
typedef __attribute__((ext_vector_type(8))) float v8f;

// ---------------------------------------------------------------------------
// WMMA f32 16x16x4 helpers (exact fp32 matmul on the matrix pipe)
// A: 16x4 (MxK): lane = M + 16*(K>=2), vgpr j -> K = 2*(lane>>4)+j
// B: 4x16 (KxN): lane = N + 16*(K>=2), vgpr j -> K = 2*(lane>>4)+j
// C/D 16x16:     lane&15 = N, vgpr r -> M = r + 8*(lane>>4)
// ---------------------------------------------------------------------------
__device__ __forceinline__ v8f wmma4(v2f a, v2f b, v8f c) {
  return __builtin_amdgcn_wmma_f32_16x16x4_f32(
      /*neg_a=*/false, a, /*neg_b=*/false, b,
      /*c_mod=*/(short)0, c, /*reuse_a=*/false, /*reuse_b=*/false);
}

__device__ __forceinline__ v2f frag_a(const float* p, int ld, int m0, int k0, int lane) {
  int m  = m0 + (lane & 15);
  int kb = k0 + ((lane >> 4) << 1);
  const float* q = p + m * ld + kb;
  v2f r; r.x = q[0]; r.y = q[1];
  return r;
}

__device__ __forceinline__ v2f frag_b(const float* p, int ld, int k0, int n0, int lane) {
  int n  = n0 + (lane & 15);
  int kb = k0 + ((lane >> 4) << 1);
  v2f r; r.x = p[kb * ld + n]; r.y = p[(kb + 1) * ld + n];
  return r;
}

__device__ __forceinline__ void frag_store(float* p, int ld, int m0, int n0, int lane, v8f d) {
  int n  = n0 + (lane & 15);
  int mb = m0 + ((lane >> 4) << 3);
#pragma unroll
  for (int r = 0; r < 8; ++r) p[(mb + r) * ld + n] = d[r];
}

#define V8Z {0.f,0.f,0.f,0.f,0.f,0.f,0.f,0.f}

// ---------------------------------------------------------------------------
// Kernel 1: input projection 63->126 with folded BN, via WMMA.
// 64 pixels per workgroup (4 waves, 1 M-tile each), N=126 (8 tiles), K=63->64.
// ---------------------------------------------------------------------------
__global__ __launch_bounds__(128) void k_proj_in(
    const float* __restrict__ x, const float* __restrict__ w_in,
    const float* __restrict__ b_in, const float* __restrict__ bn_g,
    const float* __restrict__ bn_b, const float* __restrict__ bn_m,
    const float* __restrict__ bn_v, float* __restrict__ xp)
{
  __shared__ float a_lds[64][65];    // A[m][k] = x[b][k][p0+m]
  __shared__ float w_lds[64][128];   // B[k][n] = w_in[n][k]*scale[n]
  __shared__ float bias_lds[128];
  __shared__ float d_lds[4][16][17];

  int tid  = threadIdx.x;
  int lane = tid & 31;
  int wave = tid >> 5;
  int tile = blockIdx.x;             // 0 .. 8*900-1
  int b    = tile / 900;
  int p0   = (tile % 900) * 64;

  for (int idx = tid; idx < 64 * 128; idx += 128) {
    int k = idx >> 7, n = idx & 127;
    float val = 0.f;
    if (k < 63 && n < 126) {
      float s = bn_g[n] * rsqrtf(bn_v[n] + BN_EPS);
      val = w_in[n * 63 + k] * s;
    }
    w_lds[k][n] = val;
  }
  if (tid < 128) {
    float bv = 0.f;
    if (tid < 126) {
      float s = bn_g[tid] * rsqrtf(bn_v[tid] + BN_EPS);
      bv = b_in[tid] * s + bn_b[tid] - bn_m[tid] * s;
    }
    bias_lds[tid] = bv;
  }
  for (int idx = tid; idx < 64 * 64; idx += 128) {
    int k = idx >> 6, m = idx & 63;
    a_lds[m][k] = (k < 63) ? x[(size_t)(b * 63 + k) * HW + p0 + m] : 0.f;
  }
  __syncthreads();

  int m0 = wave * 16;
  for (int nt = 0; nt < 8; ++nt) {
    int n0 = nt * 16;
    v8f acc = V8Z;
    for (int k0 = 0; k0 < 64; k0 += 4) {
      v2f av = frag_a(&a_lds[0][0], 65, m0, k0, lane);
      v2f bv = frag_b(&w_lds[0][0], 128, k0, n0, lane);
      acc = wmma4(av, bv, acc);
    }
    int nloc = lane & 15;
    int mb   = (lane >> 4) << 3;
    float bias = bias_lds[n0 + nloc];
#pragma unroll
    for (int r = 0; r < 8; ++r) d_lds[wave][mb + r][nloc] = acc[r] + bias;
    for (int idx = lane; idx < 256; idx += 32) {
      int n = idx >> 4, m = idx & 15;
      int ch = n0 + n;
      if (ch < 126)
        xp[(size_t)(b * 126 + ch) * HW + p0 + m0 + m] = d_lds[wave][m][n];
    }
  }
}

// ---------------------------------------------------------------------------
// Kernel 2: 5x5 window attention, one wave per window. blockIdx.y = group.
// Group 0: in-place (windows partition pixels). Group 1: reference's literal
// roll(-3) in, attention, roll(+2) out -> separate buffer (in-place races).
// ---------------------------------------------------------------------------
__global__ __launch_bounds__(32) void k_win(float* __restrict__ xp, float* __restrict__ y2)
{
  __shared__ float q[32][25];   // [pixel][c], K padded/zeroed to 24, rows->32
  __shared__ float v[32][25];
  __shared__ float s[32][33];

  int lane = threadIdx.x;
  int wid  = blockIdx.x;              // 0 .. 18431
  int grp  = blockIdx.y;              // 0 or 1
  int b    = wid / (NWIN * NWIN);
  int rem  = wid % (NWIN * NWIN);
  int wy   = rem / NWIN, wx = rem % NWIN;

  for (int i = lane; i < 32 * 25; i += 32) { (&q[0][0])[i] = 0.f; (&v[0][0])[i] = 0.f; }

  int chq = grp * 42;
  for (int i = lane; i < 25 * 21; i += 32) {
    int m = i / 21, c = i % 21;
    int gy = wy * WS + m / WS, gx = wx * WS + m % WS;
    if (grp == 1) { gy = (gy + 3) % HH; gx = (gx + 3) % WW; }
    size_t base = (size_t)(b * 126 + chq + c) * HW + gy * WW + gx;
    q[m][c] = xp[base];
    v[m][c] = xp[base + (size_t)21 * HW];
  }

  // S = q q^T  (25x25 in 32x32 tiles, K=21 padded to 24)
  for (int mt = 0; mt < 2; ++mt)
    for (int nt = 0; nt < 2; ++nt) {
      v8f acc = V8Z;
      for (int k0 = 0; k0 < 24; k0 += 4) {
        v2f av = frag_a(&q[0][0], 25, mt * 16, k0, lane);
        v2f bv = frag_a(&q[0][0], 25, nt * 16, k0, lane);   // B = q^T
        acc = wmma4(av, bv, acc);
      }
      frag_store(&s[0][0], 33, mt * 16, nt * 16, lane, acc);
    }

  // softmax over the 25 valid columns, one row per lane
  if (lane < 25) {
    float mx = -1e30f;
    for (int n = 0; n < 25; ++n) mx = fmaxf(mx, s[lane][n]);
    float sum = 0.f;
    for (int n = 0; n < 25; ++n) { float e = __expf(s[lane][n] - mx); s[lane][n] = e; sum += e; }
    float inv = 1.f / sum;
    for (int n = 0; n < 25; ++n) s[lane][n] *= inv;
    for (int n = 25; n < 28; ++n) s[lane][n] = 0.f;
  }

  // y = P v  (K = 25 padded to 28; v rows >= 25 are zero)
  for (int mt = 0; mt < 2; ++mt)
    for (int nt = 0; nt < 2; ++nt) {
      v8f acc = V8Z;
      for (int k0 = 0; k0 < 28; k0 += 4) {
        v2f av = frag_a(&s[0][0], 33, mt * 16, k0, lane);
        v2f bv = frag_b(&v[0][0], 25, k0, nt * 16, lane);
        acc = wmma4(av, bv, acc);
      }
      int n  = nt * 16 + (lane & 15);
      int mb = mt * 16 + ((lane >> 4) << 3);
#pragma unroll
      for (int r = 0; r < 8; ++r) {
        int m = mb + r;
        if (m < 25 && n < 21) {
          int oy = wy * WS + m / WS, ox = wx * WS + m % WS;
          if (grp == 0) {
            xp[(size_t)(b * 126 + n) * HW + oy * WW + ox] = acc[r];
          } else {
            oy = (oy + 2) % HH; ox = (ox + 2) % WW;
            y2[(size_t)(b * 21 + n) * HW + oy * WW + ox] = acc[r];
          }
        }
      }
    }
}

// ---------------------------------------------------------------------------
// Kernel 3: axial attention, one wave per (sequence, 16-query tile).
// pass 0: rows (seq over w), v from xp ch105..125, out -> vmid[b][w][h][c]
// pass 1: cols (seq over h), q from xp ch84..104, v from vmid,
//         out -> xp ch105..125 (dead v slot, disjoint from q reads)
// ---------------------------------------------------------------------------
__global__ __launch_bounds__(32) void k_axial(float* __restrict__ xp,
                                              float* __restrict__ vmid, int pass)
{
  __shared__ float q[240][25];   // [seq][c], cols 21..24 zeroed
  __shared__ float v[241][25];   // +1 row so padded-column frag reads stay in-array
  __shared__ float s[16][241];   // 16-query score strip over 240 keys

  int lane = threadIdx.x;
  int row  = blockIdx.x;         // b*240 + (h | w)
  int qt   = blockIdx.y;         // 0..14
  int b    = row / HH;
  int r    = row % HH;

  if (pass == 0) {
    for (int i = lane; i < 240 * 21; i += 32) {
      int c = i / 240, xx = i % 240;
      q[xx][c] = xp[(size_t)(b * 126 + 84 + c) * HW + r * WW + xx];
      v[xx][c] = xp[(size_t)(b * 126 + 105 + c) * HW + r * WW + xx];
    }
  } else {
    for (int i = lane; i < 240 * 21; i += 32) {
      int c = i / 240, h = i % 240;
      q[h][c] = xp[(size_t)(b * 126 + 84 + c) * HW + h * WW + r];
    }
    for (int i = lane; i < 240 * 21; i += 32) {
      int h = i / 21, c = i % 21;
      v[h][c] = vmid[((size_t)(b * 240 + r) * 240 + h) * 21 + c];
    }
  }
  for (int i = lane; i < 240 * 4; i += 32) {
    int m = i >> 2, c = 21 + (i & 3);
    q[m][c] = 0.f; v[m][c] = 0.f;
  }

  int m0 = qt * 16;
  // S strip = Q_tile K^T  (16 x 240, K = 21 padded to 24)
  for (int nt = 0; nt < 15; ++nt) {
    int n0 = nt * 16;
    v8f acc = V8Z;
    for (int k0 = 0; k0 < 24; k0 += 4) {
      v2f av = frag_a(&q[0][0], 25, m0, k0, lane);
      v2f bv = frag_a(&q[0][0], 25, n0, k0, lane);   // B = q^T
      acc = wmma4(av, bv, acc);
    }
    frag_store(&s[0][0], 241, 0, n0, lane, acc);
  }

  if (lane < 16) {
    float mx = -1e30f;
    for (int n = 0; n < 240; ++n) mx = fmaxf(mx, s[lane][n]);
    float sum = 0.f;
    for (int n = 0; n < 240; ++n) { float e = __expf(s[lane][n] - mx); s[lane][n] = e; sum += e; }
    float inv = 1.f / sum;
    for (int n = 0; n < 240; ++n) s[lane][n] *= inv;
  }

  // Y_tile = P V  (16 x 21, K = 240 -> 60 k-steps)
  for (int nt = 0; nt < 2; ++nt) {
    int n0 = nt * 16;
    v8f acc = V8Z;
    for (int k0 = 0; k0 < 240; k0 += 4) {
      v2f av = frag_a(&s[0][0], 241, 0, k0, lane);
      v2f bv = frag_b(&v[0][0], 25, k0, n0, lane);
      acc = wmma4(av, bv, acc);
    }
    int n  = n0 + (lane & 15);
    int mb = (lane >> 4) << 3;
#pragma unroll
    for (int rr = 0; rr < 8; ++rr) {
      int m = m0 + mb + rr;
      if (n < 21) {
        if (pass == 0)
          vmid[((size_t)(b * 240 + m) * 240 + r) * 21 + n] = acc[rr];
        else
          xp[(size_t)(b * 126 + 105 + n) * HW + m * WW + r] = acc[rr];
      }
    }
  }
}

// ---------------------------------------------------------------------------
// Kernel 4: output projection 63->63 via WMMA.
// y channels gathered from: xp ch0..20 (y1), y2buf (y2), xp ch105..125 (y3)
// ---------------------------------------------------------------------------
__global__ __launch_bounds__(128) void k_proj_out(
    const float* __restrict__ xp, const float* __restrict__ y2,
    const float* __restrict__ w_out, const float* __restrict__ b_out,
    float* __restrict__ out)
{
  __shared__ float a_lds[64][65];
  __shared__ float w_lds[64][64];
  __shared__ float bias_lds[64];
  __shared__ float d_lds[4][16][17];

  int tid  = threadIdx.x;
  int lane = tid & 31;
  int wave = tid >> 5;
  int tile = blockIdx.x;
  int b    = tile / 900;
  int p0   = (tile % 900) * 64;

  for (int idx = tid; idx < 64 * 64; idx += 128) {
    int k = idx >> 6, n = idx & 63;
    w_lds[k][n] = (k < 63 && n < 63) ? w_out[n * 63 + k] : 0.f;
  }
  if (tid < 64) bias_lds[tid] = (tid < 63) ? b_out[tid] : 0.f;

  for (int idx = tid; idx < 64 * 64; idx += 128) {
    int k = idx >> 6, m = idx & 63;
    float val = 0.f;
    if (k < 21)       val = xp[(size_t)(b * 126 + k)        * HW + p0 + m];
    else if (k < 42)  val = y2[(size_t)(b * 21 + (k - 21))  * HW + p0 + m];
    else if (k < 63)  val = xp[(size_t)(b * 126 + (k + 63)) * HW + p0 + m];
    a_lds[m][k] = val;
  }
  __syncthreads();

  int m0 = wave * 16;
  for (int nt = 0; nt < 4; ++nt) {
    int n0 = nt * 16;
    v8f acc = V8Z;
    for (int k0 = 0; k0 < 64; k0 += 4) {
      v2f av = frag_a(&a_lds[0][0], 65, m0, k0, lane);
      v2f bv = frag_b(&w_lds[0][0], 64, k0, n0, lane);
      acc = wmma4(av, bv, acc);
    }
    int nloc = lane & 15;
    int mb   = (lane >> 4) << 3;
    float bias = bias_lds[n0 + nloc];
#pragma unroll
    for (int r = 0; r < 8; ++r) d_lds[wave][mb + r][nloc] = acc[r] + bias;
    for (int idx = lane; idx < 256; idx += 32) {
      int n = idx >> 4, m = idx & 15;
      int ch = n0 + n;
      if (ch < 63)
        out[(size_t)(b * 63 + ch) * HW + p0 + m0 + m] = d_lds[wave][m][n];
    }
  }
}

// ---------------------------------------------------------------------------
// Host launcher. Workspace: xp (232 MB) | y2buf (38.7 MB) | vmid (38.7 MB)
// ---------------------------------------------------------------------------
extern "C" void kernel_launch(void* const* d_in, const int* in_sizes, int n_in,
                              void* d_out, int out_size, void* d_ws, size_t ws_size,
                              hipStream_t stream)
{
  (void)in_sizes; (void)n_in; (void)out_size; (void)ws_size;
  const float* x     = (const float*)d_in[0];
  const float* w_in  = (const float*)d_in[1];
  const float* b_in  = (const float*)d_in[2];
  const float* bn_g  = (const float*)d_in[3];
  const float* bn_b  = (const float*)d_in[4];
  const float* bn_m  = (const float*)d_in[5];
  const float* bn_v  = (const float*)d_in[6];
  const float* w_out = (const float*)d_in[7];
  const float* b_out = (const float*)d_in[8];
  float* out = (float*)d_out;

  float* xp   = (float*)d_ws;                        // 8*126*57600
  float* y2b  = xp  + (size_t)BB * 126 * HW;         // 8*21*57600
  float* vmid = y2b + (size_t)BB * 21 * HW;          // 8*240*240*21

  k_proj_in <<<dim3(BB * 900), 128, 0, stream>>>(x, w_in, b_in, bn_g, bn_b, bn_m, bn_v, xp);
  k_win     <<<dim3(BB * NWIN * NWIN, 2), 32, 0, stream>>>(xp, y2b);
  k_axial   <<<dim3(BB * HH, 15), 32, 0, stream>>>(xp, vmid, 0);
  k_axial   <<<dim3(BB * WW, 15), 32, 0, stream>>>(xp, vmid, 1);
  k_proj_out<<<dim3(BB * 900), 128, 0, stream>>>(xp, y2b, w_out, b_out, out);
}